// EGATLayer_17824114278571
// MI455X (gfx1250) — compile-verified
//
#include <hip/hip_runtime.h>
#include <math.h>

// Problem constants (fixed by the reference)
constexpr int N_NODES = 100000;   // divisible by 16 -> no GEMM tail
constexpr int IN_F    = 128;      // divisible by 4  -> no K tail
constexpr int CH      = 4;        // heads
constexpr int FF      = 16;       // feats/head; CH*FF = 64 output cols

typedef float v2f __attribute__((ext_vector_type(2)));
typedef float v8f __attribute__((ext_vector_type(8)));

// ---------------------------------------------------------------------------
// Kernel 1: feat = node_feat @ Wfc.T  via V_WMMA_F32_16X16X4_F32, fused with
// el = sum_f feat*attn_l, er = sum_f feat*attn_r.
// Block = 128 threads (4 waves). Block handles 16 nodes; wave w handles
// output cols [16w,16w+16) == head w.
// Fragment layouts per CDNA5 ISA 7.12.2:
//  A (16x4 f32):  lanes0-15 M=lane, VGPR0=K(0|2 by half), VGPR1=K(1|3)
//  B (4x16 f32):  lanes hold N=lane%16, VGPR0=K(0|2), VGPR1=K(1|3)
//  C/D (16x16):   VGPR r, lanes0-15 -> M=r, lanes16-31 -> M=r+8, N=lane%16
// ---------------------------------------------------------------------------
__global__ __launch_bounds__(128)
void egat_proj(const float* __restrict__ node_feat,  // [N,128]
               const float* __restrict__ Wfc,        // [64,128]
               const float* __restrict__ attn_l,     // [4,16]
               const float* __restrict__ attn_r,     // [4,16]
               float* __restrict__ feat,             // [N,64]
               float* __restrict__ el,               // [N,4]
               float* __restrict__ er,               // [N,4]
               int n_nodes)
{
    const int tile = blockIdx.x;
    const int wave = threadIdx.x >> 5;   // 0..3 == head
    const int lane = threadIdx.x & 31;
    const int half = lane >> 4;          // 0 | 1
    const int l16  = lane & 15;
    const int row0 = tile * 16;
    const int col0 = wave * 16;

    const int arow = min(row0 + l16, n_nodes - 1);
    const float* __restrict__ aptr = node_feat + arow * IN_F + half * 2;
    // B[k][j] = Wfc[col0+j][k]
    const float* __restrict__ bptr = Wfc + (col0 + l16) * IN_F + half * 2;

    v8f acc = {0.f, 0.f, 0.f, 0.f, 0.f, 0.f, 0.f, 0.f};
#pragma unroll 8
    for (int k = 0; k < IN_F; k += 4) {
        v2f a, b;
        a.x = aptr[k];     a.y = aptr[k + 1];
        b.x = bptr[k];     b.y = bptr[k + 1];
        acc = __builtin_amdgcn_wmma_f32_16x16x4_f32(
            /*neg_a=*/false, a, /*neg_b=*/false, b,
            /*c_mod=*/(short)0, acc, /*reuse_a=*/false, /*reuse_b=*/false);
    }

    const float al = attn_l[col0 + l16];   // attn_l[wave][l16]
    const float ar = attn_r[col0 + l16];

#pragma unroll
    for (int r = 0; r < 8; ++r) {
        const int m   = r + half * 8;              // output row within tile
        const int row = row0 + m;
        const float d = acc[r];
        if (row < n_nodes)
            feat[row * (CH * FF) + col0 + l16] = d;

        // 16-lane butterfly reduction (stays within each 16-lane half)
        float pl = d * al;
        float pr = d * ar;
#pragma unroll
        for (int s = 1; s < 16; s <<= 1) {
            pl += __shfl_xor(pl, s, 32);
            pr += __shfl_xor(pr, s, 32);
        }
        if (l16 == r && row < n_nodes) {
            el[row * CH + wave] = pl;
            er[row * CH + wave] = pr;
        }
    }
}

// ---------------------------------------------------------------------------
// Kernel 0: init out=0, m=-inf, s=0
// ---------------------------------------------------------------------------
__global__ void egat_fill(float* __restrict__ out, float* __restrict__ mmax,
                          float* __restrict__ ssum, int n_nodes)
{
    const int i = blockIdx.x * blockDim.x + threadIdx.x;
    if (i < n_nodes * CH * FF) out[i] = 0.f;
    if (i < n_nodes * CH) { mmax[i] = -INFINITY; ssum[i] = 0.f; }
}

// ---------------------------------------------------------------------------
// Kernel 2: per (edge, head): e = leaky_relu(el[src]+er[dst]) * (edge_feat@We.T)
//           + segment max via float atomic-max (int/uint trick, init -inf)
// ---------------------------------------------------------------------------
__global__ __launch_bounds__(256)
void egat_edge_logit(const float* __restrict__ edge_feat,  // [E,4]
                     const int* __restrict__ src,
                     const int* __restrict__ dst,
                     const float* __restrict__ We,         // [4,4]
                     const float* __restrict__ el,         // [N,4]
                     const float* __restrict__ er,         // [N,4]
                     float* __restrict__ ev,               // [E,4]
                     float* __restrict__ mmax,             // [N,4]
                     int total)
{
    const int t = blockIdx.x * 256 + threadIdx.x;
    if (t >= total) return;
    const int e = t >> 2;
    const int c = t & 3;
    const int sN = src[e];
    const int dN = dst[e];

    const float4 ef = *(const float4*)(edge_feat + e * 4);
    const float ec = We[c * 4 + 0] * ef.x + We[c * 4 + 1] * ef.y +
                     We[c * 4 + 2] * ef.z + We[c * 4 + 3] * ef.w;

    const float x  = el[sN * CH + c] + er[dN * CH + c];
    const float lr = x > 0.f ? x : 0.2f * x;
    const float v  = lr * ec;
    ev[t] = v;

    float* addr = mmax + dN * CH + c;
    if (v >= 0.f) atomicMax((int*)addr, __float_as_int(v));
    else          atomicMin((unsigned int*)addr, __float_as_uint(v));
}

// ---------------------------------------------------------------------------
// Kernel 3: ee = exp(e - m[dst]); segment sum via GLOBAL_ATOMIC_ADD_F32
// ---------------------------------------------------------------------------
__global__ __launch_bounds__(256)
void egat_expsum(const int* __restrict__ dst,
                 const float* __restrict__ mmax,
                 float* __restrict__ ev,      // in: logits, out: exp
                 float* __restrict__ ssum,    // [N,4]
                 int total)
{
    const int t = blockIdx.x * 256 + threadIdx.x;
    if (t >= total) return;
    const int e = t >> 2;
    const int c = t & 3;
    const int dN = dst[e];
    const float x = expf(ev[t] - mmax[dN * CH + c]);
    ev[t] = x;
    atomicAdd(ssum + dN * CH + c, x);
}

// ---------------------------------------------------------------------------
// Kernel 4: a = ee / s[dst]; out[dst, c, :] += feat[src, c, :] * a
// One thread per (edge, head) -> 16 float atomicAdds (L2-resident output)
// ---------------------------------------------------------------------------
__global__ __launch_bounds__(256)
void egat_aggregate(const int* __restrict__ src,
                    const int* __restrict__ dst,
                    const float* __restrict__ ev,    // exp'd logits [E,4]
                    const float* __restrict__ ssum,  // [N,4]
                    const float* __restrict__ feat,  // [N,64]
                    float* __restrict__ out,         // [N,64]
                    int total)
{
    const int t = blockIdx.x * 256 + threadIdx.x;
    if (t >= total) return;
    const int e = t >> 2;
    const int c = t & 3;
    const int sN = src[e];
    const int dN = dst[e];
    const float a = ev[t] / ssum[dN * CH + c];

    const float4* __restrict__ fsrc = (const float4*)(feat + sN * (CH * FF) + c * FF);
    float* __restrict__ o = out + dN * (CH * FF) + c * FF;
#pragma unroll
    for (int q = 0; q < 4; ++q) {
        const float4 f = fsrc[q];
        atomicAdd(o + q * 4 + 0, f.x * a);
        atomicAdd(o + q * 4 + 1, f.y * a);
        atomicAdd(o + q * 4 + 2, f.z * a);
        atomicAdd(o + q * 4 + 3, f.w * a);
    }
}

// ---------------------------------------------------------------------------
extern "C" void kernel_launch(void* const* d_in, const int* in_sizes, int n_in,
                              void* d_out, int out_size, void* d_ws, size_t ws_size,
                              hipStream_t stream)
{
    const float* node_feat = (const float*)d_in[0];  // [N,128]
    const float* edge_feat = (const float*)d_in[1];  // [E,4]
    const int*   src       = (const int*)d_in[2];    // [E]
    const int*   dst       = (const int*)d_in[3];    // [E]
    const float* Wfc       = (const float*)d_in[4];  // [64,128]
    const float* We        = (const float*)d_in[5];  // [4,4]
    const float* attn_l    = (const float*)d_in[6];  // [4,16]
    const float* attn_r    = (const float*)d_in[7];  // [4,16]
    float*       out       = (float*)d_out;          // [N,64]

    const int n = in_sizes[0] / IN_F;   // 100000
    const int E = in_sizes[2];          // 3200000

    // Workspace layout (floats): feat | el | er | mmax | ssum | ev
    float* feat = (float*)d_ws;                    // n*64
    float* el   = feat + (size_t)n * (CH * FF);    // n*4
    float* er   = el   + (size_t)n * CH;           // n*4
    float* mmax = er   + (size_t)n * CH;           // n*4
    float* ssum = mmax + (size_t)n * CH;           // n*4
    float* ev   = ssum + (size_t)n * CH;           // E*4

    // 0) init output / max / sum
    {
        const int total = n * CH * FF;
        egat_fill<<<(total + 255) / 256, 256, 0, stream>>>(out, mmax, ssum, n);
    }
    // 1) WMMA projection + el/er
    {
        const int tiles = (n + 15) / 16;   // 6250
        egat_proj<<<tiles, 128, 0, stream>>>(node_feat, Wfc, attn_l, attn_r,
                                             feat, el, er, n);
    }
    const int totalEC = E * CH;            // 12.8M
    const int gEdge   = (totalEC + 255) / 256;
    // 2) edge logits + segment max
    egat_edge_logit<<<gEdge, 256, 0, stream>>>(edge_feat, src, dst, We, el, er,
                                               ev, mmax, totalEC);
    // 3) exp + segment sum
    egat_expsum<<<gEdge, 256, 0, stream>>>(dst, mmax, ev, ssum, totalEC);
    // 4) normalize + scatter-add aggregation
    egat_aggregate<<<gEdge, 256, 0, stream>>>(src, dst, ev, ssum, feat, out,
                                              totalEC);
}